// StaticGIBGAT_55052890800195
// MI455X (gfx1250) — compile-verified
//
#include <hip/hip_runtime.h>
#include <hip/hip_bf16.h>

// ---------------- problem constants (match reference) ----------------
#define NN    10000
#define E0N   100000
#define EN    (E0N + NN)     // 110000 edges incl. self loops
#define INF_  512
#define HIDN  256
#define OUTN  10
#define H1N   4
#define H2N   1
#define KMIX  5
#define ONN   (2 * HIDN)     // 512
#define D1N   (H1N * HIDN)   // 1024
#define D2N   (H2N * HIDN)   // 256
#define LOG2PI_F 1.8378770664093453f

typedef float v2f __attribute__((ext_vector_type(2)));
typedef float v4f __attribute__((ext_vector_type(4)));
typedef float v8f __attribute__((ext_vector_type(8)));

// ---------------- helpers ----------------
__device__ __forceinline__ float waveReduceSum(float v) {
    #pragma unroll
    for (int o = 16; o > 0; o >>= 1) v += __shfl_xor(v, o, 32);
    return v;
}

// monotonic-int trick for float atomic max (init with -inf)
__device__ __forceinline__ void atomicMaxF(float* addr, float v) {
    if (v >= 0.0f) atomicMax((int*)addr, __float_as_int(v));
    else           atomicMin((unsigned int*)addr, (unsigned int)__float_as_int(v));
}

__device__ __forceinline__ void edge_sd(const int* __restrict__ ei, int e, int& s, int& d) {
    if (e < E0N) { s = ei[e]; d = ei[E0N + e]; }
    else         { s = e - E0N; d = e - E0N; }
}

__device__ __forceinline__ float eluf(float x)      { return x > 0.0f ? x : expm1f(x); }
__device__ __forceinline__ float softplusf(float x) { return x > 20.0f ? x : log1pf(expf(x)); }

// ---------------- generic fill ----------------
__global__ void fill_kernel(float* __restrict__ p, long n, float v) {
    long i = (long)blockIdx.x * blockDim.x + threadIdx.x;
    long stride = (long)gridDim.x * blockDim.x;
    for (; i < n; i += stride) p[i] = v;
}

// =====================================================================
// WMMA f32 GEMM, LDS double-buffered: C[M,N] = A[M,K] @ B[K,N], row-major.
// Block = 256 threads = 8 waves; block tile = 128x64 (wave tile 16x64,
// 4 accumulators). K consumed in KC=32 chunks; the 32x64 B chunk is staged
// cooperatively into LDS once per block and double-buffered so the global
// loads for chunk c+1 overlap the 32 WMMAs of chunk c.
// B LDS layout: pair-row interleave [k/2][n][2] with stride PRS=160 floats
// -> each B fragment is one aligned ds_load_b64, and the +hi pair-row
// offset lands lanes 16-31 on the opposite 32-bank half (160 % 64 == 32).
// Requires K%32==0, N%64==0 (true here); M tail handled by clamping.
// =====================================================================
#define GKC   32
#define GBN   64
#define PRS   160                 // pair-row stride (floats)
#define LDSB  (16 * PRS)          // floats per buffer (KC/2 pair rows)

__global__ __launch_bounds__(256)
void gemm_wmma_f32_tiled(const float* __restrict__ A, const float* __restrict__ B,
                         float* __restrict__ C, int M, int N, int K) {
    __shared__ float Bs[2 * LDSB];

    const int NG   = N >> 6;
    const int mb   = blockIdx.x / NG;
    const int gc0  = (blockIdx.x % NG) << 6;
    const int t    = threadIdx.x;
    const int wave = t >> 5;
    const int lane = t & 31;
    const int hi   = lane >> 4;
    const int l16  = lane & 15;

    const int MT  = (M + 15) >> 4;
    const int tm  = mb * 8 + wave;
    const int tmc = tm < MT ? tm : MT - 1;        // clamp so all waves hit barriers
    const float* Arow = A + (size_t)(tmc * 16 + l16) * K + 2 * hi;

    v8f acc0 = {0,0,0,0,0,0,0,0};
    v8f acc1 = {0,0,0,0,0,0,0,0};
    v8f acc2 = {0,0,0,0,0,0,0,0};
    v8f acc3 = {0,0,0,0,0,0,0,0};

    v2f aCur[8], aNext[8];
    v4f breg[2];

    // ---- prologue: chunk 0 ----
    #pragma unroll
    for (int j = 0; j < 8; ++j) aCur[j] = *(const v2f*)(Arow + 4 * j);
    #pragma unroll
    for (int i = 0; i < 2; ++i) {
        int idx4 = t + i * 256;                   // 512 float4 = 32x64 chunk
        int row = idx4 >> 4, col4 = idx4 & 15;
        breg[i] = *(const v4f*)(B + (size_t)row * N + gc0 + col4 * 4);
    }
    #pragma unroll
    for (int i = 0; i < 2; ++i) {
        int idx4 = t + i * 256;
        int row = idx4 >> 4, c0 = (idx4 & 15) * 4;
        #pragma unroll
        for (int j = 0; j < 4; ++j)
            Bs[(row >> 1) * PRS + (c0 + j) * 2 + (row & 1)] = breg[i][j];
    }
    __syncthreads();

    const int nch = K / GKC;
    for (int c = 0; c < nch; ++c) {
        // ---- issue global prefetch for chunk c+1 (overlaps WMMAs below) ----
        if (c + 1 < nch) {
            const int kc = (c + 1) * GKC;
            #pragma unroll
            for (int j = 0; j < 8; ++j) aNext[j] = *(const v2f*)(Arow + kc + 4 * j);
            #pragma unroll
            for (int i = 0; i < 2; ++i) {
                int idx4 = t + i * 256;
                int row = idx4 >> 4, col4 = idx4 & 15;
                breg[i] = *(const v4f*)(B + (size_t)(kc + row) * N + gc0 + col4 * 4);
            }
        }
        // ---- compute 8 k-steps from LDS buffer c&1, b-frags pipelined ----
        const float* Bl = Bs + (c & 1) * LDSB + hi * PRS + 2 * l16;
        v2f b0 = *(const v2f*)(Bl + 0);
        v2f b1 = *(const v2f*)(Bl + 32);
        v2f b2 = *(const v2f*)(Bl + 64);
        v2f b3 = *(const v2f*)(Bl + 96);
        #pragma unroll
        for (int j = 0; j < 8; ++j) {
            v2f n0, n1, n2, n3;
            if (j < 7) {
                const float* Bn = Bl + (j + 1) * 2 * PRS;
                n0 = *(const v2f*)(Bn + 0);
                n1 = *(const v2f*)(Bn + 32);
                n2 = *(const v2f*)(Bn + 64);
                n3 = *(const v2f*)(Bn + 96);
            }
            acc0 = __builtin_amdgcn_wmma_f32_16x16x4_f32(false, aCur[j], false, b0, (short)0, acc0, false, false);
            acc1 = __builtin_amdgcn_wmma_f32_16x16x4_f32(false, aCur[j], false, b1, (short)0, acc1, false, false);
            acc2 = __builtin_amdgcn_wmma_f32_16x16x4_f32(false, aCur[j], false, b2, (short)0, acc2, false, false);
            acc3 = __builtin_amdgcn_wmma_f32_16x16x4_f32(false, aCur[j], false, b3, (short)0, acc3, false, false);
            b0 = n0; b1 = n1; b2 = n2; b3 = n3;
        }
        // ---- stage chunk c+1 into the other buffer ----
        if (c + 1 < nch) {
            #pragma unroll
            for (int i = 0; i < 2; ++i) {
                int idx4 = t + i * 256;
                int row = idx4 >> 4, c0 = (idx4 & 15) * 4;
                float* dst = Bs + ((c + 1) & 1) * LDSB;
                #pragma unroll
                for (int j = 0; j < 4; ++j)
                    dst[(row >> 1) * PRS + (c0 + j) * 2 + (row & 1)] = breg[i][j];
            }
            __syncthreads();
            #pragma unroll
            for (int j = 0; j < 8; ++j) aCur[j] = aNext[j];
        }
    }

    // ---- store 16x64 strip (C/D layout: VGPR r -> row r + 8*hi) ----
    if (tm < MT) {
        const int row0 = tm * 16 + hi * 8;
        #pragma unroll
        for (int r = 0; r < 8; ++r) {
            float* crow = C + (size_t)(row0 + r) * N + gc0 + l16;
            crow[0]  = acc0[r];
            crow[16] = acc1[r];
            crow[32] = acc2[r];
            crow[48] = acc3[r];
        }
    }
}

// ---------------- attention scores: one wave per (node, head) ----------------
__global__ __launch_bounds__(256)
void att_scores_kernel(const float* __restrict__ xh, const float* __restrict__ att,
                       float* __restrict__ si, float* __restrict__ sj, int NH, int H) {
    const int w = blockIdx.x * (blockDim.x >> 5) + (threadIdx.x >> 5);
    if (w >= NH) return;
    const int lane = threadIdx.x & 31;
    const int n = w / H, h = w % H;
    const float* xr = xh + (size_t)n * H * ONN + (size_t)h * ONN;
    const float* al = att + (size_t)h * 2 * ONN;
    const float* ar = al + ONN;
    float a = 0.0f, b = 0.0f;
    for (int c = lane; c < ONN; c += 32) { float v = xr[c]; a += v * al[c]; b += v * ar[c]; }
    a = waveReduceSum(a); b = waveReduceSum(b);
    if (lane == 0) { si[w] = a; sj[w] = b; }
}

// ---------------- in-degree (incl. self loops) ----------------
__global__ void edge_deg_kernel(const int* __restrict__ ei, float* __restrict__ deg) {
    int e = blockIdx.x * blockDim.x + threadIdx.x;
    if (e >= EN) return;
    int s, d; edge_sd(ei, e, s, d);
    atomicAdd(&deg[d], 1.0f);
}

// ---------------- leaky-relu logits + segment max ----------------
__global__ void edge_logit_kernel(const int* __restrict__ ei, const float* __restrict__ si,
                                  const float* __restrict__ sj, float* __restrict__ ebuf,
                                  float* __restrict__ segmax, int H) {
    int e = blockIdx.x * blockDim.x + threadIdx.x;
    if (e >= EN) return;
    int s, d; edge_sd(ei, e, s, d);
    for (int h = 0; h < H; ++h) {
        float lg = si[d * H + h] + sj[s * H + h];
        lg = lg > 0.0f ? lg : 0.2f * lg;
        ebuf[(size_t)e * H + h] = lg;
        atomicMaxF(&segmax[d * H + h], lg);
    }
}

// ---------------- exp(logit - segmax) + segment sum ----------------
__global__ void edge_exp_kernel(const int* __restrict__ ei, float* __restrict__ ebuf,
                                const float* __restrict__ segmax, float* __restrict__ segsum, int H) {
    int e = blockIdx.x * blockDim.x + threadIdx.x;
    if (e >= EN) return;
    int s, d; edge_sd(ei, e, s, d);
    for (int h = 0; h < H; ++h) {
        float ex = expf(ebuf[(size_t)e * H + h] - segmax[d * H + h]);
        ebuf[(size_t)e * H + h] = ex;
        atomicAdd(&segsum[d * H + h], ex);
    }
}

// ---------------- alpha, KL vs 1/deg prior, gumbel logits + segment max ----------------
__global__ void edge_alpha_kl_gumbel_kernel(const int* __restrict__ ei, const float* __restrict__ ebuf,
                                            const float* __restrict__ segsum, const float* __restrict__ deg,
                                            const float* __restrict__ gum, float* __restrict__ ebuf2,
                                            float* __restrict__ segmax2, float* __restrict__ kl_acc, int H) {
    int e = blockIdx.x * blockDim.x + threadIdx.x;
    float klc = 0.0f;
    if (e < EN) {
        int s, d; edge_sd(ei, e, s, d);
        float prior = 1.0f / deg[d];
        for (int h = 0; h < H; ++h) {
            float alpha = ebuf[(size_t)e * H + h] / (segsum[d * H + h] + 1e-16f);
            klc += alpha * (logf(alpha + 1e-12f) - logf(prior + 1e-12f));
            float lg2 = logf(alpha + 1e-16f) + gum[(size_t)e * H + h];
            ebuf2[(size_t)e * H + h] = lg2;
            atomicMaxF(&segmax2[d * H + h], lg2);
        }
    }
    klc = waveReduceSum(klc);
    if ((threadIdx.x & 31) == 0) atomicAdd(kl_acc, klc);
}

// ---------------- scatter aggregation: one wave per (edge, head) ----------------
__global__ __launch_bounds__(256)
void edge_aggregate_kernel(const int* __restrict__ ei, const float* __restrict__ xh,
                           const float* __restrict__ ebuf2, const float* __restrict__ segsum2,
                           float* __restrict__ agg, int H) {
    const int p = blockIdx.x * (blockDim.x >> 5) + (threadIdx.x >> 5);
    if (p >= EN * H) return;
    const int lane = threadIdx.x & 31;
    const int e = p / H, h = p % H;
    int s, d; edge_sd(ei, e, s, d);
    const float w = ebuf2[(size_t)e * H + h] / (segsum2[d * H + h] + 1e-16f);
    const float* xr = xh + (size_t)s * H * ONN + (size_t)h * ONN;
    float* orow     = agg + (size_t)d * H * ONN + (size_t)h * ONN;
    for (int c = lane; c < ONN; c += 32) atomicAdd(&orow[c], xr[c] * w);
}

// ---------------- reparameterize + ELBO (log_q - mixture prior) + elu ----------------
__global__ void node_latent_kernel(const float* __restrict__ agg, const float* __restrict__ eps,
                                   const float* __restrict__ pi, const float* __restrict__ mu,
                                   const float* __restrict__ logsig, float* __restrict__ hout,
                                   float* __restrict__ ixz_acc, int H, int D, long total) {
    long t = (long)blockIdx.x * blockDim.x + threadIdx.x;
    float contrib = 0.0f;
    if (t < total) {
        int n = (int)(t / D), d = (int)(t % D);
        int h = d / HIDN, c = d % HIDN;
        const float* arow = agg + (size_t)n * H * ONN + (size_t)h * ONN;
        float mean = arow[c];
        float sd   = softplusf(arow[HIDN + c]) + 1e-10f;
        float ep   = eps[t];
        float z    = mean + sd * ep;
        hout[t] = eluf(z);
        float lq = -0.5f * ep * ep - logf(sd) - 0.5f * LOG2PI_F;
        float pmx = pi[0];
        #pragma unroll
        for (int k = 1; k < KMIX; ++k) pmx = fmaxf(pmx, pi[k]);
        float pse = 0.0f;
        #pragma unroll
        for (int k = 0; k < KMIX; ++k) pse += expf(pi[k] - pmx);
        float plse = pmx + logf(pse);
        float cmx = -3.0e38f, comp[KMIX];
        #pragma unroll
        for (int k = 0; k < KMIX; ++k) {
            float ls  = logsig[(size_t)d * KMIX + k];
            float sg  = expf(ls);
            float dv  = (z - mu[(size_t)d * KMIX + k]) / sg;
            comp[k]   = -0.5f * dv * dv - ls - 0.5f * LOG2PI_F + (pi[k] - plse);
            cmx = fmaxf(cmx, comp[k]);
        }
        float cse = 0.0f;
        #pragma unroll
        for (int k = 0; k < KMIX; ++k) cse += expf(comp[k] - cmx);
        float plp = cmx + logf(cse);
        contrib = lq - plp;
    }
    contrib = waveReduceSum(contrib);
    if ((threadIdx.x & 31) == 0) atomicAdd(ixz_acc, contrib);
}

// ---------------- final FC: logits = h2 @ Wfc + bfc ----------------
__global__ void fc_kernel(const float* __restrict__ h2, const float* __restrict__ Wfc,
                          const float* __restrict__ bfc, float* __restrict__ out) {
    int t = blockIdx.x * blockDim.x + threadIdx.x;
    if (t >= NN * OUTN) return;
    int n = t / OUTN, o = t % OUTN;
    float acc = bfc[o];
    const float* hr = h2 + (size_t)n * D2N;
    for (int d = 0; d < D2N; ++d) acc += hr[d] * Wfc[(size_t)d * OUTN + o];
    out[t] = acc;
}

// ---------------- scalar outputs ----------------
__global__ void finalize_kernel(const float* __restrict__ accum, float* __restrict__ out) {
    if (threadIdx.x == 0 && blockIdx.x == 0) {
        float kl  = accum[0] / (float)(EN * H1N) + accum[2] / (float)(EN * H2N);
        float ixz = (accum[1] + accum[3]) / (float)NN;
        out[(size_t)NN * OUTN]     = kl;
        out[(size_t)NN * OUTN + 1] = ixz;
    }
}

// =====================================================================
extern "C" void kernel_launch(void* const* d_in, const int* in_sizes, int n_in,
                              void* d_out, int out_size, void* d_ws, size_t ws_size,
                              hipStream_t stream) {
    const float* x       = (const float*)d_in[0];
    const int*   ei      = (const int*)  d_in[1];
    const float* W1      = (const float*)d_in[2];
    const float* att1    = (const float*)d_in[3];
    const float* pi1     = (const float*)d_in[4];
    const float* mu1     = (const float*)d_in[5];
    const float* logsig1 = (const float*)d_in[6];
    const float* eps1    = (const float*)d_in[7];
    const float* g1      = (const float*)d_in[8];
    const float* W2      = (const float*)d_in[9];
    const float* att2    = (const float*)d_in[10];
    const float* pi2     = (const float*)d_in[11];
    const float* mu2     = (const float*)d_in[12];
    const float* logsig2 = (const float*)d_in[13];
    const float* eps2    = (const float*)d_in[14];
    const float* g2      = (const float*)d_in[15];
    const float* Wfc     = (const float*)d_in[16];
    const float* bfc     = (const float*)d_in[17];
    float* out = (float*)d_out;

    // ---- workspace layout (floats), regions reused across layers ----
    float* ws = (float*)d_ws;
    const size_t BIG = (size_t)NN * (H1N * ONN);            // 20.48M floats
    float* xh1  = ws;                                        // layer1 xh [N, 2048]
    float* agg  = ws + BIG;                                  // aggregation (reused l2)
    float* h1   = ws + 2 * BIG;                              // elu(Z1) [N, 1024]
    float* sc   = h1 + (size_t)NN * D1N;
    float* deg  = sc;              sc += NN;
    float* si   = sc;              sc += (size_t)NN * H1N;
    float* sj   = sc;              sc += (size_t)NN * H1N;
    float* m1   = sc;              sc += (size_t)NN * H1N;
    float* s1   = sc;              sc += (size_t)NN * H1N;
    float* m2   = sc;              sc += (size_t)NN * H1N;
    float* s2   = sc;              sc += (size_t)NN * H1N;
    float* e1   = sc;              sc += (size_t)EN * H1N;
    float* e2   = sc;              sc += (size_t)EN * H1N;
    float* accum = sc;             sc += 4;                  // kl1, ixz1, kl2, ixz2
    float* xh2 = ws;                                         // layer2 xh [N, 512]
    float* h2  = ws + (size_t)NN * (H2N * ONN);              // elu(Z2) [N, 256]

    const int TB = 256;
    const int EB = (EN + TB - 1) / TB;
    const int MT = (NN + 15) / 16;        // 625
    const int MB = (MT + 7) / 8;          // 79

    // ---- per-call init ----
    fill_kernel<<<1024, TB, 0, stream>>>(deg, NN, 0.0f);
    fill_kernel<<<1, 32, 0, stream>>>(accum, 4, 0.0f);
    edge_deg_kernel<<<EB, TB, 0, stream>>>(ei, deg);

    // ======================= Layer 1 (H = 4) =======================
    gemm_wmma_f32_tiled<<<MB * (2048 / 64), TB, 0, stream>>>(x, W1, xh1, NN, H1N * ONN, INF_);
    att_scores_kernel<<<(NN * H1N) / 8, TB, 0, stream>>>(xh1, att1, si, sj, NN * H1N, H1N);
    fill_kernel<<<1024, TB, 0, stream>>>(m1, (long)NN * H1N, -INFINITY);
    fill_kernel<<<1024, TB, 0, stream>>>(s1, (long)NN * H1N, 0.0f);
    fill_kernel<<<1024, TB, 0, stream>>>(m2, (long)NN * H1N, -INFINITY);
    fill_kernel<<<1024, TB, 0, stream>>>(s2, (long)NN * H1N, 0.0f);
    edge_logit_kernel<<<EB, TB, 0, stream>>>(ei, si, sj, e1, m1, H1N);
    edge_exp_kernel<<<EB, TB, 0, stream>>>(ei, e1, m1, s1, H1N);
    edge_alpha_kl_gumbel_kernel<<<EB, TB, 0, stream>>>(ei, e1, s1, deg, g1, e2, m2, accum + 0, H1N);
    edge_exp_kernel<<<EB, TB, 0, stream>>>(ei, e2, m2, s2, H1N);
    fill_kernel<<<2048, TB, 0, stream>>>(agg, (long)NN * H1N * ONN, 0.0f);
    edge_aggregate_kernel<<<(EN * H1N + 7) / 8, TB, 0, stream>>>(ei, xh1, e2, s2, agg, H1N);
    node_latent_kernel<<<(int)(((long)NN * D1N + TB - 1) / TB), TB, 0, stream>>>(
        agg, eps1, pi1, mu1, logsig1, h1, accum + 1, H1N, D1N, (long)NN * D1N);

    // ======================= Layer 2 (H = 1) =======================
    gemm_wmma_f32_tiled<<<MB * (512 / 64), TB, 0, stream>>>(h1, W2, xh2, NN, H2N * ONN, D1N);
    att_scores_kernel<<<(NN * H2N + 7) / 8, TB, 0, stream>>>(xh2, att2, si, sj, NN * H2N, H2N);
    fill_kernel<<<1024, TB, 0, stream>>>(m1, (long)NN * H2N, -INFINITY);
    fill_kernel<<<1024, TB, 0, stream>>>(s1, (long)NN * H2N, 0.0f);
    fill_kernel<<<1024, TB, 0, stream>>>(m2, (long)NN * H2N, -INFINITY);
    fill_kernel<<<1024, TB, 0, stream>>>(s2, (long)NN * H2N, 0.0f);
    edge_logit_kernel<<<EB, TB, 0, stream>>>(ei, si, sj, e1, m1, H2N);
    edge_exp_kernel<<<EB, TB, 0, stream>>>(ei, e1, m1, s1, H2N);
    edge_alpha_kl_gumbel_kernel<<<EB, TB, 0, stream>>>(ei, e1, s1, deg, g2, e2, m2, accum + 2, H2N);
    edge_exp_kernel<<<EB, TB, 0, stream>>>(ei, e2, m2, s2, H2N);
    fill_kernel<<<2048, TB, 0, stream>>>(agg, (long)NN * H2N * ONN, 0.0f);
    edge_aggregate_kernel<<<(EN * H2N + 7) / 8, TB, 0, stream>>>(ei, xh2, e2, s2, agg, H2N);
    node_latent_kernel<<<(int)(((long)NN * D2N + TB - 1) / TB), TB, 0, stream>>>(
        agg, eps2, pi2, mu2, logsig2, h2, accum + 3, H2N, D2N, (long)NN * D2N);

    // ======================= Head =======================
    fc_kernel<<<(NN * OUTN + TB - 1) / TB, TB, 0, stream>>>(h2, Wfc, bfc, out);
    finalize_kernel<<<1, 32, 0, stream>>>(accum, out);
}